// CTCLoss_2800318677325
// MI455X (gfx1250) — compile-verified
//
#include <hip/hip_runtime.h>
#include <hip/hip_bf16.h>
#include <cstdint>

// Problem constants (from reference setup_inputs)
#define BB 1024
#define TT 256
#define CC 128
#define LL 32
#define BLANK 127      // C-1
#define RING 8         // async LDS ring depth (8 rows x 512B = 4KB in flight)
#define NEGINF (-1e30f)
#define EPSF (1e-7f)

// CDNA5 async global->LDS path (ASYNCcnt). Probe-confirmed present on this
// toolchain (round-1 diagnostic showed the builtin's signature); guarded so
// the file still compiles on toolchains without it.
#if defined(__HIP_DEVICE_COMPILE__) && \
    __has_builtin(__builtin_amdgcn_global_load_async_to_lds_b128)
#define HAVE_ASYNC 1
#else
#define HAVE_ASYNC 0
#endif

#if HAVE_ASYNC
// Builtin signature (from hipcc diagnostic): param0 is a pointer to
// 'int __attribute__((vector_size(16)))' in the global (__device__ / AS1)
// address space; param1 is the LDS (AS3) side.
typedef int v4i_t __attribute__((vector_size(4 * sizeof(int))));
typedef __attribute__((address_space(1))) v4i_t g_v4i;
typedef __attribute__((address_space(3))) v4i_t l_v4i;
#endif

template <int N>
__device__ __forceinline__ void wait_async() {
#if HAVE_ASYNC
#if __has_builtin(__builtin_amdgcn_s_wait_asynccnt)
  __builtin_amdgcn_s_wait_asynccnt(N);
#else
  asm volatile("s_wait_asynccnt %0" ::"n"(N) : "memory");
#endif
#endif
}

// Copy one 512B probability row (32 lanes x 16B) global -> LDS, tracked by
// ASYNCcnt (in-order completion per the CDNA5 ISA).
__device__ __forceinline__ void issue_row(const float* __restrict__ gsrc,
                                          float* ldst, int lane) {
#if HAVE_ASYNC
  __builtin_amdgcn_global_load_async_to_lds_b128(
      (g_v4i*)(gsrc + lane * 4), (l_v4i*)(ldst + lane * 4),
      /*imm offset=*/0, /*cpol=*/0);
#else
  reinterpret_cast<float4*>(ldst)[lane] =
      reinterpret_cast<const float4*>(gsrc)[lane];
#endif
}

__device__ __forceinline__ float lse3(float x0, float x1, float x2) {
  float m = fmaxf(fmaxf(x0, x1), x2);
  float s = __expf(x0 - m) + __expf(x1 - m) + __expf(x2 - m);
  return m + __logf(s);
}

// One DP timestep: 65 CTC states as {a0 (s=lane), a1 (s=lane+32), a64}.
// All s-1 / s-2 shifts are wave32 cross-lane shuffles -> zero barriers.
__device__ __forceinline__ void ctc_step(const float* __restrict__ row, int lane,
                                         int e0, int e1, bool blk0, bool blk1,
                                         float& a0, float& a1, float& a64) {
  float lp0 = __logf(row[e0] + EPSF);
  float lp1 = __logf(row[e1] + EPSF);
  float lpB = __logf(row[BLANK] + EPSF);  // state 64 (LDS broadcast read)

  float top0a = __shfl(a0, 31);  // state 31
  float top0b = __shfl(a0, 30);  // state 30
  float top1a = __shfl(a1, 31);  // state 63
  // shift by 1: alpha[s-1]
  float u0 = __shfl_up(a0, 1);
  if (lane == 0) u0 = NEGINF;
  float u1 = __shfl_up(a1, 1);
  if (lane == 0) u1 = top0a;
  // shift by 2: alpha[s-2]
  float v0 = __shfl_up(a0, 2);
  if (lane < 2) v0 = NEGINF;
  float v1 = __shfl_up(a1, 2);
  if (lane == 0) v1 = top0b;
  else if (lane == 1) v1 = top0a;

  float n0 = fmaxf(lse3(a0, u0, blk0 ? NEGINF : v0) + lp0, NEGINF);
  float n1 = fmaxf(lse3(a1, u1, blk1 ? NEGINF : v1) + lp1, NEGINF);
  float n64 = fmaxf(lse3(a64, top1a, NEGINF) + lpB, NEGINF);
  a0 = n0;
  a1 = n1;
  a64 = n64;
}

// One wave32 per batch element; 8-deep async LDS ring streams y_pred rows
// ahead of the sequential T-scan so HBM latency stays off the critical path.
__global__ __launch_bounds__(32) void ctc_cost_kernel(
    const float* __restrict__ y_pred,  // [B,T,C] probabilities
    const int* __restrict__ y_true,    // [B,L]
    float* __restrict__ out) {         // [B]
  __shared__ float rows[RING][CC];     // 4KB ring of probability rows

  const int b = blockIdx.x;
  const int lane = threadIdx.x;
  const float* yp = y_pred + (size_t)b * (TT * CC);
  const int* lab = y_true + b * LL;

  // Extended label sequence ext[s]: blank at even s, label[s>>1] at odd s.
  const int s0 = lane;               // state in 0..31
  const int s1 = lane + 32;          // state in 32..63
  const bool odd = (lane & 1) != 0;  // s0 and s1 share parity
  int e0 = BLANK, e1 = BLANK;
  bool blk0 = true, blk1 = true;  // skip(s-2) blocked?
  if (odd) {
    const int i0 = s0 >> 1;
    const int i1 = s1 >> 1;
    const int l0 = lab[i0];
    const int l1 = lab[i1];
    e0 = l0;
    e1 = l1;
    blk0 = (s0 == 1) ? true : (l0 == lab[i0 - 1]);
    blk1 = (l1 == lab[i1 - 1]);  // s1 >= 33 always has a valid i1-1
  }
  // state 64 is even (blank) -> its s-2 skip is always blocked.

  // Prime the pipeline: rows 0..RING-1 in flight.
#pragma unroll
  for (int r = 0; r < RING; ++r) issue_row(yp + r * CC, rows[r], lane);
  wait_async<RING - 1>();  // in-order completion => row 0 landed

  // alpha(t=0): only states 0,1 live; e0 is BLANK on lane0, lab[0] on lane1.
  float a0 = NEGINF, a1 = NEGINF, a64 = NEGINF;
  {
    float lp = __logf(rows[0][e0] + EPSF);
    if (lane < 2) a0 = lp;
  }

  // Steady state: refill the slot freed last iteration (row t-1's) with row
  // t+RING-1, then wait until the oldest in-flight row (row t) has landed.
  for (int t = 1; t <= TT - RING; ++t) {
    const int r = t + RING - 1;
    issue_row(yp + r * CC, rows[r & (RING - 1)], lane);
    wait_async<RING - 1>();
    ctc_step(rows[t & (RING - 1)], lane, e0, e1, blk0, blk1, a0, a1, a64);
  }

  // Tail: rows TT-RING+1 .. TT-1 are all in flight; one drain, then
  // branch-free steps.
  wait_async<0>();
  for (int t = TT - RING + 1; t < TT; ++t) {
    ctc_step(rows[t & (RING - 1)], lane, e0, e1, blk0, blk1, a0, a1, a64);
  }

  // loss = -logaddexp(alpha[S-1], alpha[S-2])
  float a63 = __shfl(a1, 31);
  float m = fmaxf(a64, a63);
  float nll = -(m + __logf(__expf(a64 - m) + __expf(a63 - m)));
  if (lane == 0) out[b] = nll;
}

extern "C" void kernel_launch(void* const* d_in, const int* in_sizes, int n_in,
                              void* d_out, int out_size, void* d_ws,
                              size_t ws_size, hipStream_t stream) {
  (void)in_sizes;
  (void)n_in;
  (void)d_ws;
  (void)ws_size;
  const int* y_true = (const int*)d_in[0];      // [B,L] int
  const float* y_pred = (const float*)d_in[1];  // [B,T,C] float32
  float* out = (float*)d_out;                   // [B,1] float32

  dim3 grid((unsigned)out_size);  // one single-wave block per batch element
  dim3 block(32);
  ctc_cost_kernel<<<grid, block, 0, stream>>>(y_pred, y_true, out);
}